// NonLocalAttention_28209345200523
// MI455X (gfx1250) — compile-verified
//
#include <hip/hip_runtime.h>

// ---------------- problem constants ----------------
#define DIMC   96          // channels
#define C3     288         // 3*DIM
#define HEADS  3
#define CH     32          // channels per head
#define WSZ    8           // window size
#define NTOK   64          // WSZ*WSZ tokens per window
#define SHIFT  4
#define HIMG   264
#define WIMG   264
#define HW     69696       // HIMG*WIMG
#define TSTRIDE 50         // tile origins 0,50,100,150,200
#define TK     64          // tile size
#define NT1D   5
#define NTILES 25

typedef float v2f __attribute__((ext_vector_type(2)));
typedef float v8f __attribute__((ext_vector_type(8)));

// D = A(16x4,f32) * B(4x16,f32) + C  -- CDNA5 fp32 WMMA
__device__ __forceinline__ v8f wmma4(v2f a, v2f b, v8f c) {
  return __builtin_amdgcn_wmma_f32_16x16x4_f32(false, a, false, b, (short)0, c,
                                               false, false);
}

// ---------------- K1: qkv = qkv_w @ roll(x) ----------------
// grid: (HW/16, 3), block 32. Each wave: 96 output rows x 16 pixels.
__global__ __launch_bounds__(32)
void k_qkv_gemm(const float* __restrict__ x, const float* __restrict__ w,
                float* __restrict__ qkv0) {
  const int lane = threadIdx.x;
  const int n0 = blockIdx.x * 16;
  const int m0 = blockIdx.y * 96;
  const int ln = lane & 15;
  const int kb = (lane >> 4) * 2;
  const int p  = n0 + ln;
  const int y  = p / WIMG, xx = p % WIMG;
  const int sp = ((y + SHIFT) % HIMG) * WIMG + ((xx + SHIFT) % WIMG);

  v8f acc[6] = {};
  for (int k0 = 0; k0 < DIMC; k0 += 4) {
    v2f b;
    b.x = x[(size_t)(k0 + kb)     * HW + sp];
    b.y = x[(size_t)(k0 + kb + 1) * HW + sp];
#pragma unroll
    for (int t = 0; t < 6; ++t) {
      const float* wr = w + (size_t)(m0 + t * 16 + ln) * DIMC + k0 + kb;
      v2f a; a.x = wr[0]; a.y = wr[1];
      acc[t] = wmma4(a, b, acc[t]);
    }
  }
#pragma unroll
  for (int t = 0; t < 6; ++t)
#pragma unroll
    for (int r = 0; r < 8; ++r) {
      int row = m0 + t * 16 + r + 8 * (lane >> 4);
      qkv0[(size_t)row * HW + p] = acc[t][r];
    }
}

// ---------------- K2: 3x3 depthwise conv, zero pad ----------------
__global__ void k_dwconv(const float* __restrict__ qkv0,
                         const float* __restrict__ dw,
                         float* __restrict__ qkv) {
  int idx = blockIdx.x * blockDim.x + threadIdx.x;
  if (idx >= C3 * HW) return;
  int o = idx / HW, p = idx % HW;
  int y = p / WIMG, x = p % WIMG;
  const float* wv = dw + o * 9;
  const float* in = qkv0 + (size_t)o * HW;
  float s = 0.f;
#pragma unroll
  for (int ky = 0; ky < 3; ++ky) {
    int yy = y + ky - 1;
    if (yy < 0 || yy >= HIMG) continue;
#pragma unroll
    for (int kx = 0; kx < 3; ++kx) {
      int xs = x + kx - 1;
      if (xs < 0 || xs >= WIMG) continue;
      s += in[yy * WIMG + xs] * wv[ky * 3 + kx];
    }
  }
  qkv[idx] = s;
}

// ---------------- K3: per (tile,window,head) channel attention ----------------
// grid: NTILES*64*HEADS, block 32 (one wave).
__global__ __launch_bounds__(32)
void k_attn(const float* __restrict__ qkv, const float* __restrict__ temperature,
            float* __restrict__ out_tiles) {
  __shared__ float lq[CH * NTOK], lk[CH * NTOK], lv[CH * NTOK];
  __shared__ float lattn[CH * CH];
  __shared__ float rq[CH], rk[CH];

  const int lane = threadIdx.x;
  int unit = blockIdx.x;
  int head = unit % HEADS;
  int win  = (unit / HEADS) & 63;
  int tile = unit / (HEADS * 64);
  int ti = (tile / NT1D) * TSTRIDE;
  int tj = (tile % NT1D) * TSTRIDE;
  int wi = win >> 3, wj = win & 7;
  int by = ti + wi * WSZ, bx = tj + wj * WSZ;

  const size_t qoff = (size_t)(head * CH) * HW;
  for (int idx = lane; idx < CH * NTOK; idx += 32) {
    int c = idx >> 6, n = idx & 63;
    int sp = (by + (n >> 3)) * WIMG + bx + (n & 7);
    lq[idx] = qkv[qoff + (size_t)c * HW + sp];
    lk[idx] = qkv[qoff + (size_t)(DIMC + c) * HW + sp];
    lv[idx] = qkv[qoff + (size_t)(2 * DIMC + c) * HW + sp];
  }
  __syncthreads();

  { // reciprocal L2 norms per channel row (lane == row)
    float sq = 0.f, sk = 0.f;
    for (int n = 0; n < NTOK; ++n) {
      float a = lq[lane * NTOK + n]; sq += a * a;
      float b = lk[lane * NTOK + n]; sk += b * b;
    }
    rq[lane] = 1.0f / fmaxf(sqrtf(sq), 1e-12f);
    rk[lane] = 1.0f / fmaxf(sqrtf(sk), 1e-12f);
  }
  __syncthreads();

  const int ln = lane & 15, kb = (lane >> 4) * 2;

  // attn(32x32) = Q(32x64) * K(32x64)^T
  v8f acc[2][2] = {};
  for (int k0 = 0; k0 < NTOK; k0 += 4) {
    v2f af[2], bf[2];
#pragma unroll
    for (int t = 0; t < 2; ++t) {
      af[t].x = lq[(t * 16 + ln) * NTOK + k0 + kb];
      af[t].y = lq[(t * 16 + ln) * NTOK + k0 + kb + 1];
      bf[t].x = lk[(t * 16 + ln) * NTOK + k0 + kb];
      bf[t].y = lk[(t * 16 + ln) * NTOK + k0 + kb + 1];
    }
#pragma unroll
    for (int i = 0; i < 2; ++i)
#pragma unroll
      for (int j = 0; j < 2; ++j) acc[i][j] = wmma4(af[i], bf[j], acc[i][j]);
  }

  float tmp = temperature[head];
#pragma unroll
  for (int i = 0; i < 2; ++i)
#pragma unroll
    for (int j = 0; j < 2; ++j)
#pragma unroll
      for (int r = 0; r < 8; ++r) {
        int row = i * 16 + r + 8 * (lane >> 4);
        int col = j * 16 + ln;
        float v = acc[i][j][r] * rq[row] * rk[col] * tmp;
        lattn[row * CH + col] = fmaxf(v, 0.f);
      }
  __syncthreads();

  // out(32x64) = attn(32x32) * V(32x64)
  v8f o[2][4] = {};
  for (int k0 = 0; k0 < CH; k0 += 4) {
    v2f af[2], bf[4];
#pragma unroll
    for (int i = 0; i < 2; ++i) {
      af[i].x = lattn[(i * 16 + ln) * CH + k0 + kb];
      af[i].y = lattn[(i * 16 + ln) * CH + k0 + kb + 1];
    }
#pragma unroll
    for (int j = 0; j < 4; ++j) {
      int n = j * 16 + ln;
      bf[j].x = lv[(k0 + kb)     * NTOK + n];
      bf[j].y = lv[(k0 + kb + 1) * NTOK + n];
    }
#pragma unroll
    for (int i = 0; i < 2; ++i)
#pragma unroll
      for (int j = 0; j < 4; ++j) o[i][j] = wmma4(af[i], bf[j], o[i][j]);
  }

  const size_t tbase = ((size_t)tile * DIMC + head * CH) * (TK * TK);
#pragma unroll
  for (int i = 0; i < 2; ++i)
#pragma unroll
    for (int j = 0; j < 4; ++j)
#pragma unroll
      for (int r = 0; r < 8; ++r) {
        int row = i * 16 + r + 8 * (lane >> 4);
        int col = j * 16 + ln;
        int py = wi * WSZ + (col >> 3), px = wj * WSZ + (col & 7);
        out_tiles[tbase + (size_t)row * (TK * TK) + py * TK + px] = o[i][j][r];
      }
}

// ---------------- K4a: overlap gather + average ----------------
__global__ void k_gather(const float* __restrict__ out_tiles,
                         float* __restrict__ preds) {
  int idx = blockIdx.x * blockDim.x + threadIdx.x;
  if (idx >= DIMC * HW) return;
  int c = idx / HW, p = idx % HW;
  int y = p / WIMG, x = p % WIMG;
  int rl[2], cl[2]; int rn = 0, cn = 0;
#pragma unroll
  for (int t = 0; t < NT1D; ++t) {
    int o = t * TSTRIDE;
    if (y >= o && y < o + TK) rl[rn++] = t;
    if (x >= o && x < o + TK) cl[cn++] = t;
  }
  float s = 0.f;
  for (int a = 0; a < rn; ++a)
    for (int b = 0; b < cn; ++b) {
      int tile = rl[a] * NT1D + cl[b];
      s += out_tiles[((size_t)tile * DIMC + c) * (TK * TK) +
                     (y - rl[a] * TSTRIDE) * TK + (x - cl[b] * TSTRIDE)];
    }
  preds[idx] = s / (float)(rn * cn);
}

// ---------------- K4b: out = roll_back(proj_w @ preds) ----------------
__global__ __launch_bounds__(32)
void k_proj(const float* __restrict__ preds, const float* __restrict__ pw,
            float* __restrict__ out) {
  const int lane = threadIdx.x;
  const int n0 = blockIdx.x * 16;
  const int ln = lane & 15, kb = (lane >> 4) * 2;
  const int p = n0 + ln;

  v8f acc[6] = {};
  for (int k0 = 0; k0 < DIMC; k0 += 4) {
    v2f b;
    b.x = preds[(size_t)(k0 + kb)     * HW + p];
    b.y = preds[(size_t)(k0 + kb + 1) * HW + p];
#pragma unroll
    for (int t = 0; t < 6; ++t) {
      const float* wr = pw + (size_t)(t * 16 + ln) * DIMC + k0 + kb;
      v2f a; a.x = wr[0]; a.y = wr[1];
      acc[t] = wmma4(a, b, acc[t]);
    }
  }
  int y = p / WIMG, x = p % WIMG;
  int dst = ((y + SHIFT) % HIMG) * WIMG + ((x + SHIFT) % WIMG);
#pragma unroll
  for (int t = 0; t < 6; ++t)
#pragma unroll
    for (int r = 0; r < 8; ++r) {
      int row = t * 16 + r + 8 * (lane >> 4);
      out[(size_t)row * HW + dst] = acc[t][r];
    }
}

// ---------------- launch ----------------
extern "C" void kernel_launch(void* const* d_in, const int* in_sizes, int n_in,
                              void* d_out, int out_size, void* d_ws,
                              size_t ws_size, hipStream_t stream) {
  (void)in_sizes; (void)n_in; (void)out_size; (void)ws_size;
  const float* x    = (const float*)d_in[0];
  const float* temp = (const float*)d_in[1];
  const float* qw   = (const float*)d_in[2];
  const float* dw   = (const float*)d_in[3];
  const float* pw   = (const float*)d_in[4];
  float* out = (float*)d_out;

  float* qkv0 = (float*)d_ws;                       // 288*HW floats
  float* qkv  = qkv0 + (size_t)C3 * HW;             // 288*HW floats
  // qkv0 region is dead after k_dwconv -> reuse it:
  float* out_tiles = qkv0;                          // 25*96*4096 = 9,830,400 f
  float* preds = qkv0 + (size_t)NTILES * DIMC * (TK * TK); // +6,690,816 f (fits)

  k_qkv_gemm<<<dim3(HW / 16, 3), 32, 0, stream>>>(x, qw, qkv0);

  {
    int n = C3 * HW;
    k_dwconv<<<(n + 255) / 256, 256, 0, stream>>>(qkv0, dw, qkv);
  }

  k_attn<<<NTILES * 64 * HEADS, 32, 0, stream>>>(qkv, temp, out_tiles);

  {
    int n = DIMC * HW;
    k_gather<<<(n + 255) / 256, 256, 0, stream>>>(out_tiles, preds);
  }

  k_proj<<<HW / 16, 32, 0, stream>>>(preds, pw, out);
}